// GraphConv_69707319214514
// MI455X (gfx1250) — compile-verified
//
#include <hip/hip_runtime.h>
#include <hip/hip_bf16.h>

typedef __attribute__((ext_vector_type(2))) float v2f;
typedef __attribute__((ext_vector_type(8))) float v8f;

#define IN_F  128
#define OUT_F 128

// ---------------- degree / normalization ----------------

__global__ void gcn_init_deg(float* __restrict__ deg, int n) {
    int i = blockIdx.x * blockDim.x + threadIdx.x;
    if (i < n) deg[i] = 1.0f;  // self loop
}

__global__ void gcn_count_deg(const long long* __restrict__ dst,
                              float* __restrict__ deg, long long E) {
    long long e = blockIdx.x * (long long)blockDim.x + threadIdx.x;
    if (e < E) atomicAdd(&deg[dst[e]], 1.0f);
}

__global__ void gcn_finalize_dinv(float* __restrict__ deg, int n) {
    int i = blockIdx.x * blockDim.x + threadIdx.x;
    if (i < n) deg[i] = rsqrtf(fmaxf(deg[i], 1e-12f));
}

// ---------------- dense GEMM h = x @ W via V_WMMA_F32_16X16X4_F32 ----------------
// One wave per 16-row strip; the wave accumulates the full 16x128 output strip
// in 8 fp32 16x16 accumulators, stepping K in chunks of 4.
//
// Per ISA 7.12.2 (32-bit operands, wave32):
//   A (16x4):  lane L, vgpr v ->  row m = L&15,        k = v + 2*(L>>4)
//   B (4x16):  lane L, vgpr v ->  col n = L&15,        k = v + 2*(L>>4)
//   C (16x16): lane L, vgpr v ->  row m = v + 8*(L>>4), col n = L&15

__global__ void gcn_gemm_wmma(const float* __restrict__ X,
                              const float* __restrict__ W,
                              float* __restrict__ H, int n) {
    const int wave  = threadIdx.x >> 5;
    const int lane  = threadIdx.x & 31;
    const int strip = blockIdx.x * (blockDim.x >> 5) + wave;
    const int row0  = strip * 16;
    if (row0 >= n) return;  // wave-uniform: EXEC stays all-ones for WMMA

    const int m     = lane & 15;
    const int khalf = (lane >> 4) << 1;  // 0 or 2

    v8f acc[8] = {};

    const float* __restrict__ xrow = X + (size_t)(row0 + m) * IN_F;

    #pragma unroll 4
    for (int k = 0; k < IN_F; k += 4) {
        v2f a;
        a.x = xrow[k + khalf];
        a.y = xrow[k + khalf + 1];
        const float* __restrict__ wrow = W + (size_t)(k + khalf) * OUT_F + m;
        #pragma unroll
        for (int j = 0; j < 8; ++j) {
            v2f b;
            b.x = wrow[j * 16];              // W[k+khalf  ][j*16 + m]
            b.y = wrow[j * 16 + OUT_F];      // W[k+khalf+1][j*16 + m]
            acc[j] = __builtin_amdgcn_wmma_f32_16x16x4_f32(
                false, a, false, b, (short)0, acc[j], false, false);
        }
    }

    const int mbase = (lane >> 4) << 3;  // 0 or 8
    #pragma unroll
    for (int j = 0; j < 8; ++j) {
        #pragma unroll
        for (int v = 0; v < 8; ++v) {
            H[(size_t)(row0 + mbase + v) * OUT_F + j * 16 + m] = acc[j][v];
        }
    }
}

// ---------------- self-loop + bias init: out = bias + h * dinv^2 ----------------

__global__ void gcn_init_out(const float* __restrict__ H,
                             const float* __restrict__ dinv,
                             const float* __restrict__ bias,
                             float* __restrict__ out, long long total) {
    long long t = blockIdx.x * (long long)blockDim.x + threadIdx.x;
    if (t >= total) return;
    long long node = t >> 7;       // / 128
    int f = (int)(t & 127);
    float di = dinv[node];
    out[t] = bias[f] + H[t] * di * di;
}

// ---------------- edge scatter: out[dst] += h[src] * dinv[src]*dinv[dst] ----------------
// One wave per edge; lane handles 4 features (float4 gather from L2-resident h).

__global__ void gcn_scatter(const long long* __restrict__ src,
                            const long long* __restrict__ dst,
                            const float* __restrict__ H,
                            const float* __restrict__ dinv,
                            float* __restrict__ out, long long E) {
    long long t = blockIdx.x * (long long)blockDim.x + threadIdx.x;
    long long e = t >> 5;
    if (e >= E) return;
    const int f = (int)(t & 31) << 2;        // lane*4 features
    const long long s = src[e];              // wave-uniform -> scalar loads
    const long long d = dst[e];
    const float nrm = dinv[s] * dinv[d];
    const float4 hv = *(const float4*)(H + s * OUT_F + f);
    float* __restrict__ op = out + d * OUT_F + f;
    atomicAdd(op + 0, hv.x * nrm);
    atomicAdd(op + 1, hv.y * nrm);
    atomicAdd(op + 2, hv.z * nrm);
    atomicAdd(op + 3, hv.w * nrm);
}

// ---------------- host-side launch ----------------

extern "C" void kernel_launch(void* const* d_in, const int* in_sizes, int n_in,
                              void* d_out, int out_size, void* d_ws, size_t ws_size,
                              hipStream_t stream) {
    const float*     x     = (const float*)d_in[0];
    const long long* index = (const long long*)d_in[1];   // int64 [2, E]
    const float*     w     = (const float*)d_in[2];
    const float*     bias  = (const float*)d_in[3];
    float*           out   = (float*)d_out;

    const int       N = in_sizes[0] / IN_F;       // 50000
    const long long E = in_sizes[1] / 2;          // 625000
    const long long* srcs = index;
    const long long* dsts = index + E;

    // workspace: [dinv: N floats][pad to 256B][H: N*128 floats]
    float* dinv = (float*)d_ws;
    size_t dinv_bytes = ((size_t)N * sizeof(float) + 255) & ~(size_t)255;
    float* H = (float*)((char*)d_ws + dinv_bytes);

    const int T = 256;

    gcn_init_deg<<<(N + T - 1) / T, T, 0, stream>>>(dinv, N);
    gcn_count_deg<<<(int)((E + T - 1) / T), T, 0, stream>>>(dsts, dinv, E);
    gcn_finalize_dinv<<<(N + T - 1) / T, T, 0, stream>>>(dinv, N);

    const int strips = (N + 15) / 16;             // 3125
    const int wavesPerBlock = T / 32;             // 8
    gcn_gemm_wmma<<<(strips + wavesPerBlock - 1) / wavesPerBlock, T, 0, stream>>>(x, w, H, N);

    const long long total = (long long)N * OUT_F;
    gcn_init_out<<<(int)((total + T - 1) / T), T, 0, stream>>>(H, dinv, bias, out, total);

    const long long scatterThreads = E * 32;
    gcn_scatter<<<(int)((scatterThreads + T - 1) / T), T, 0, stream>>>(srcs, dsts, H, dinv, out, E);
}